// SoftMedianConv_51505247814286
// MI455X (gfx1250) — compile-verified
//
#include <hip/hip_runtime.h>
#include <hip/hip_bf16.h>

typedef __attribute__((ext_vector_type(2))) float v2f;
typedef __attribute__((ext_vector_type(8))) float v8f;

#define NN     50000
#define KDEG   17
#define DIN    128
#define DOUT   64
// LDS layout for B: for each k-pair row r2 = r/2, element pair
// (W[2*r2][c], W[2*r2+1][c]) stored at lw[r2*LDS_S2 + 2*c .. +1].
// LDS_S2 = 160 floats => adjacent k-pairs differ by 32 banks (mod 64),
// so the two half-waves of a B-fragment ds_load_b64 are conflict-free.
#define LDS_S2 160

// ---------------------------------------------------------------------------
// Kernel 1: xp = x @ W via V_WMMA_F32_16X16X4_F32 (fp32-exact WMMA path).
// Block = 128 threads (4 waves); block computes a 64x64 output tile.
// Weight (128x64 fp32) staged pair-interleaved in LDS via async-to-LDS DMA,
// so every B fragment is a single aligned ds_load_b64 into a VGPR pair
// (eliminates the v_dual_mov repacking seen in the previous build).
// ---------------------------------------------------------------------------
__global__ void __launch_bounds__(128)
softmed_gemm_kernel(const float* __restrict__ x,
                    const float* __restrict__ wgt,
                    float* __restrict__ xp) {
    __shared__ float lw[(DIN / 2) * LDS_S2];   // 40 KB

    // Async-stage weight into LDS with the pair-interleave swizzle.
    // Each lane moves 64 elements; ASYNCcnt tracks completion.
    for (int i = threadIdx.x; i < DIN * DOUT; i += 128) {
        const int r = i >> 6;
        const int c = i & 63;
        const unsigned ldsa =
            (unsigned)(uintptr_t)&lw[(r >> 1) * LDS_S2 + 2 * c + (r & 1)];
        const unsigned long long ga = (unsigned long long)(uintptr_t)(wgt + i);
        asm volatile("global_load_async_to_lds_b32 %0, %1, off"
                     :: "v"(ldsa), "v"(ga) : "memory");
    }
    asm volatile("s_wait_asynccnt 0" ::: "memory");
    __syncthreads();

    const int wave = threadIdx.x >> 5;
    const int lane = threadIdx.x & 31;
    const int half = lane >> 4;     // 0: lanes 0-15, 1: lanes 16-31
    const int l15  = lane & 15;
    const int rowBase = blockIdx.x * 64 + wave * 16;

    // clamp A-fragment row so the partial last block keeps EXEC all-ones
    int arow = rowBase + l15;
    if (arow >= NN) arow = NN - 1;
    const float* __restrict__ xrow = x + (long)arow * DIN;

    v8f acc0 = {}, acc1 = {}, acc2 = {}, acc3 = {};

    for (int k = 0; k < DIN; k += 4) {
        // A: VGPR0 = K = k + 2*half, VGPR1 = K+1  -> one global_load_b64
        v2f a = *(const v2f*)(xrow + k + 2 * half);

        // B fragments: one ds_load_b64 each, already in WMMA layout
        const float* __restrict__ bb = &lw[((k >> 1) + half) * LDS_S2];
        v2f b0 = *(const v2f*)(bb + 2 * (0 * 16 + l15));
        v2f b1 = *(const v2f*)(bb + 2 * (1 * 16 + l15));
        v2f b2 = *(const v2f*)(bb + 2 * (2 * 16 + l15));
        v2f b3 = *(const v2f*)(bb + 2 * (3 * 16 + l15));

        acc0 = __builtin_amdgcn_wmma_f32_16x16x4_f32(false, a, false, b0, (short)0, acc0, false, false);
        acc1 = __builtin_amdgcn_wmma_f32_16x16x4_f32(false, a, false, b1, (short)0, acc1, false, false);
        acc2 = __builtin_amdgcn_wmma_f32_16x16x4_f32(false, a, false, b2, (short)0, acc2, false, false);
        acc3 = __builtin_amdgcn_wmma_f32_16x16x4_f32(false, a, false, b3, (short)0, acc3, false, false);
    }

    // D layout: VGPR j holds row (j + 8*half), column = l15 (16x16 f32 C/D map)
    #pragma unroll
    for (int j = 0; j < 8; ++j) {
        int m = rowBase + j + half * 8;
        if (m < NN) {
            float* o = xp + (long)m * DOUT + l15;
            o[0 * 16] = acc0[j];
            o[1 * 16] = acc1[j];
            o[2 * 16] = acc2[j];
            o[3 * 16] = acc3[j];
        }
    }
}

// ---------------------------------------------------------------------------
// Kernel 2: per-(node,dim) weighted median over 17 neighbors, then
// out = row_sum * med + bias. One thread per output element; 64 consecutive
// threads share a node so gathers are coalesced and L2-resident (xp = 12.8MB
// << 192MB L2). Pairwise rank accumulation exploits antisymmetry: one compare
// per unordered pair (136 instead of 272); v/w/below all stay in registers.
// Tie-break (v[i]<=v[j] for i<j) matches jnp.argsort's stable order.
// ---------------------------------------------------------------------------
__global__ void __launch_bounds__(256)
softmed_median_kernel(const float* __restrict__ xp,
                      const int*   __restrict__ col,
                      const float* __restrict__ ew,
                      const float* __restrict__ bias,
                      float*       __restrict__ out) {
    const int t = blockIdx.x * 256 + threadIdx.x;
    const int n = t >> 6;      // node
    const int d = t & 63;      // dim
    if (n >= NN) return;

    const int e0 = n * KDEG;

    float v[KDEG], w[KDEG];
    float total = 0.0f;
    #pragma unroll
    for (int i = 0; i < KDEG; ++i) {
        const int c = col[e0 + i];
        v[i] = xp[c * DOUT + d];
        w[i] = ew[e0 + i];
        total += w[i];
    }
    const float target = 0.5f * total;

    // below[j] = sum of w_i over (v_i < v_j) || (v_i == v_j && i < j)
    float below[KDEG];
    #pragma unroll
    for (int j = 0; j < KDEG; ++j) below[j] = 0.0f;

    #pragma unroll
    for (int j = 1; j < KDEG; ++j) {
        #pragma unroll
        for (int i = 0; i < j; ++i) {
            const bool le = (v[i] <= v[j]);   // stable order for i<j
            below[j] += le ? w[i] : 0.0f;
            below[i] += le ? 0.0f : w[j];
        }
    }

    // exactly one j satisfies the crossing condition
    float med = v[0];
    #pragma unroll
    for (int j = 0; j < KDEG; ++j) {
        const bool ismed = (below[j] < target) && (below[j] + w[j] >= target);
        med = ismed ? v[j] : med;
    }

    out[t] = total * med + bias[d];
}

extern "C" void kernel_launch(void* const* d_in, const int* in_sizes, int n_in,
                              void* d_out, int out_size, void* d_ws, size_t ws_size,
                              hipStream_t stream) {
    const float* x          = (const float*)d_in[0];   // [NN, DIN]
    const int*   edge_index = (const int*)  d_in[1];   // [2, E]
    const float* ew         = (const float*)d_in[2];   // [E]
    const float* wgt        = (const float*)d_in[3];   // [DIN, DOUT]
    const float* bias       = (const float*)d_in[4];   // [DOUT]
    float*       out        = (float*)d_out;           // [NN, DOUT]

    float* xp = (float*)d_ws;                          // [NN, DOUT] scratch (12.8 MB)

    const int E = NN * KDEG;
    const int* col = edge_index + E;                   // second row of edge_index

    const int gemm_blocks = (NN + 63) / 64;            // 782 (last block partial)
    softmed_gemm_kernel<<<gemm_blocks, 128, 0, stream>>>(x, wgt, xp);

    const int med_threads = NN * DOUT;                 // 3.2M
    const int med_blocks  = (med_threads + 255) / 256; // 12500
    softmed_median_kernel<<<med_blocks, 256, 0, stream>>>(xp, col, ew, bias, out);
}